// GCNNet_30442728194281
// MI455X (gfx1250) — compile-verified
//
#include <hip/hip_runtime.h>

#define N_NODES 50000
#define N_EDGES 600000
#define HIDDEN 128
#define NUM_GRAPHS 128
#define ROW_TILES (N_NODES / 16)   // 3125 exactly

typedef __attribute__((ext_vector_type(2))) float v2f;
typedef __attribute__((ext_vector_type(8))) float v8f;

// ---------------------------------------------------------------------------
// Degree count: deg[c] += 1 for every edge target (self-loop +1 added in dinv)
// ---------------------------------------------------------------------------
__global__ void k_degree(const int* __restrict__ col, float* __restrict__ deg) {
    int e = blockIdx.x * blockDim.x + threadIdx.x;
    if (e < N_EDGES) atomicAdd(&deg[col[e]], 1.0f);
}

__global__ void k_dinv(float* __restrict__ dinv) {
    int n = blockIdx.x * blockDim.x + threadIdx.x;
    if (n < N_NODES) dinv[n] = rsqrtf(dinv[n] + 1.0f);  // +1 self-loop
}

// ---------------------------------------------------------------------------
// MLP layer 1: h0[n][j] = relu(x[n] * W1[j] + b1[j])   (x is [N,1] -> outer)
// ---------------------------------------------------------------------------
__global__ void k_mlp1(const float* __restrict__ x, const float* __restrict__ W1,
                       const float* __restrict__ b1, float* __restrict__ out) {
    int t = blockIdx.x * blockDim.x + threadIdx.x;
    if (t >= N_NODES * HIDDEN) return;
    int n = t >> 7, j = t & 127;
    out[t] = fmaxf(fmaf(x[n], W1[j], b1[j]), 0.0f);
}

// ---------------------------------------------------------------------------
// WMMA fp32 GEMM: D[N x ncols] = A[N x 128] * W[128 x ncols] (+bias)(+relu)
// One wave per 16x16 output tile; K=128 = 32 steps of v_wmma_f32_16x16x4_f32.
// ---------------------------------------------------------------------------
__global__ __launch_bounds__(128) void
k_gemm(const float* __restrict__ A, const float* __restrict__ W,
       const float* __restrict__ bias, float* __restrict__ D,
       int ncols, int do_relu) {
    const int lane    = threadIdx.x;        // 0..31
    const int wave    = threadIdx.y;        // 0..3
    const int ct      = blockIdx.x;         // column tile
    const int rt      = blockIdx.y * 4 + wave;
    if (rt >= ROW_TILES) return;            // wave-uniform exit (EXEC stays full)
    const int halfsel = lane >> 4;          // 0: K pair {0,1}, 1: K pair {2,3}
    const int lidx    = lane & 15;
    const int col     = ct * 16 + lidx;

    // Preload B: weight tile is reused by all 3125 row tiles -> keep in VGPRs.
    v2f breg[32];
#pragma unroll
    for (int kk = 0; kk < 32; ++kk) {
        const int k0 = kk * 4 + halfsel * 2;
        breg[kk].x = W[k0 * ncols + col];
        breg[kk].y = W[(k0 + 1) * ncols + col];
    }

    // C/D layout: element i lives at row (i + 8*halfsel), col lidx -> bias is
    // column-only, broadcast into all 8 accumulator elements.
    const float bv = bias ? bias[col] : 0.0f;
    v8f acc;
#pragma unroll
    for (int i = 0; i < 8; ++i) acc[i] = bv;

    const float* arow = A + (rt * 16 + lidx) * HIDDEN;
#pragma unroll
    for (int kk = 0; kk < 32; ++kk) {
        const int k0 = kk * 4 + halfsel * 2;
        v2f a;
        a.x = arow[k0];
        a.y = arow[k0 + 1];
        acc = __builtin_amdgcn_wmma_f32_16x16x4_f32(
            /*neg_a=*/false, a, /*neg_b=*/false, breg[kk],
            /*c_mod=*/(short)0, acc, /*reuse_a=*/false, /*reuse_b=*/false);
    }

    float* drow = D + (rt * 16 + halfsel * 8) * ncols + col;
#pragma unroll
    for (int i = 0; i < 8; ++i) {
        float v = acc[i];
        if (do_relu) v = fmaxf(v, 0.0f);
        drow[i * ncols] = v;
    }
}

// ---------------------------------------------------------------------------
// hnew[n][j] = bg[j]  (scatter target init with layer bias)
// ---------------------------------------------------------------------------
__global__ void k_bias_init(const float* __restrict__ bg, float* __restrict__ h) {
    int t = blockIdx.x * blockDim.x + threadIdx.x;
    if (t < N_NODES * HIDDEN) h[t] = bg[t & 127];
}

// ---------------------------------------------------------------------------
// Edge scatter: one wave per edge (incl. N_NODES trailing self-loops).
// lane handles 4 consecutive floats: hnew[c][:] += m[r][:] * dinv[r]*dinv[c]
// ---------------------------------------------------------------------------
__global__ __launch_bounds__(256) void
k_scatter(const int* __restrict__ row, const int* __restrict__ col,
          const float* __restrict__ dinv, const float* __restrict__ m,
          float* __restrict__ hnew) {
    const int e = blockIdx.x * 8 + threadIdx.y;
    if (e >= N_EDGES + N_NODES) return;
    int r, c;
    if (e < N_EDGES) { r = row[e]; c = col[e]; }
    else             { r = e - N_EDGES; c = r; }           // self-loop
    const float w = dinv[r] * dinv[c];
    const int lane = threadIdx.x;
    const float4 mv = ((const float4*)(m + (size_t)r * HIDDEN))[lane];
    float* dst = hnew + (size_t)c * HIDDEN + lane * 4;
    atomicAdd(dst + 0, mv.x * w);
    atomicAdd(dst + 1, mv.y * w);
    atomicAdd(dst + 2, mv.z * w);
    atomicAdd(dst + 3, mv.w * w);
}

__global__ void k_relu(float* __restrict__ h) {
    int t = blockIdx.x * blockDim.x + threadIdx.x;
    if (t < N_NODES * HIDDEN) h[t] = fmaxf(h[t], 0.0f);
}

// ---------------------------------------------------------------------------
// Head: per node dot(h3[n], W4) + b4, wave32 shuffle reduce, add into graph.
// ---------------------------------------------------------------------------
__global__ __launch_bounds__(256) void
k_head(const float* __restrict__ h3, const float* __restrict__ W4,
       const float* __restrict__ b4, const int* __restrict__ batch,
       float* __restrict__ out) {
    const int n = blockIdx.x * 8 + threadIdx.y;
    if (n >= N_NODES) return;
    const int lane = threadIdx.x;
    const float2 v = ((const float2*)(h3 + (size_t)n * 64))[lane];
    float s = v.x * W4[2 * lane] + v.y * W4[2 * lane + 1];
#pragma unroll
    for (int off = 16; off > 0; off >>= 1) s += __shfl_down(s, off, 32);
    if (lane == 0) atomicAdd(&out[batch[n]], s + b4[0]);
}

// ---------------------------------------------------------------------------
extern "C" void kernel_launch(void* const* d_in, const int* in_sizes, int n_in,
                              void* d_out, int out_size, void* d_ws, size_t ws_size,
                              hipStream_t stream) {
    const float* x     = (const float*)d_in[0];
    // d_in[1] = pos : unused by the reference
    const int*   eidx  = (const int*)d_in[2];
    const int*   batch = (const int*)d_in[3];
    const float* W1    = (const float*)d_in[4];
    const float* b1    = (const float*)d_in[5];
    const float* W2    = (const float*)d_in[6];
    const float* b2    = (const float*)d_in[7];
    const float* Wg    = (const float*)d_in[8];
    const float* bg    = (const float*)d_in[9];
    const float* W3    = (const float*)d_in[10];
    const float* b3    = (const float*)d_in[11];
    const float* W4    = (const float*)d_in[12];
    const float* b4    = (const float*)d_in[13];
    float* out = (float*)d_out;

    const int* row = eidx;              // edge_index[0]
    const int* col = eidx + N_EDGES;    // edge_index[1]

    float* buf_h = (float*)d_ws;                         // [N,128]
    float* buf_m = buf_h + (size_t)N_NODES * HIDDEN;     // [N,128]
    float* dinv  = buf_m + (size_t)N_NODES * HIDDEN;     // [N] (deg, then dinv)
    float* h3    = dinv + N_NODES;                       // [N,64]

    const int EW = N_NODES * HIDDEN;
    hipMemsetAsync(dinv, 0, N_NODES * sizeof(float), stream);
    hipMemsetAsync(out, 0, NUM_GRAPHS * sizeof(float), stream);

    // gcn_norm
    k_degree<<<(N_EDGES + 255) / 256, 256, 0, stream>>>(col, dinv);
    k_dinv<<<(N_NODES + 255) / 256, 256, 0, stream>>>(dinv);

    // node MLP: h = relu(x@W1+b1) @ W2 + b2
    k_mlp1<<<(EW + 255) / 256, 256, 0, stream>>>(x, W1, b1, buf_m);
    const dim3 gblk(32, 4);
    const dim3 ggrid128(HIDDEN / 16, (ROW_TILES + 3) / 4);
    k_gemm<<<ggrid128, gblk, 0, stream>>>(buf_m, W2, b2, buf_h, HIDDEN, 0);

    // 3 GCN layers
    for (int i = 0; i < 3; ++i) {
        k_gemm<<<ggrid128, gblk, 0, stream>>>(
            buf_h, Wg + (size_t)i * HIDDEN * HIDDEN, nullptr, buf_m, HIDDEN, 0);
        k_bias_init<<<(EW + 255) / 256, 256, 0, stream>>>(bg + i * HIDDEN, buf_h);
        k_scatter<<<(N_EDGES + N_NODES + 7) / 8, dim3(32, 8), 0, stream>>>(
            row, col, dinv, buf_m, buf_h);
        k_relu<<<(EW + 255) / 256, 256, 0, stream>>>(buf_h);
    }

    // readout MLP: relu(h@W3+b3) @ W4 + b4 -> segment_sum over batch
    const dim3 ggrid64(64 / 16, (ROW_TILES + 3) / 4);
    k_gemm<<<ggrid64, gblk, 0, stream>>>(buf_h, W3, b3, h3, 64, 1);
    k_head<<<(N_NODES + 7) / 8, dim3(32, 8), 0, stream>>>(h3, W4, b4, batch, out);
}